// WindingEmbedding_1735166787895
// MI455X (gfx1250) — compile-verified
//
#include <hip/hip_runtime.h>
#include <math.h>

typedef float v2f __attribute__((ext_vector_type(2)));
typedef float v8f __attribute__((ext_vector_type(8)));

#define VOCAB_SZ 50257
#define EMB_D    1024
#define NFEAT    16      // 2 * NUM_WINDINGS
#define LN_EPS   1e-5f

// ---------------------------------------------------------------------------
// Prologue: precompute LayerNorm stats helpers into workspace.
//   ws[0..255]   : G[i][j]  = sum_d W[d][i]*W[d][j]   (W^T W, 16x16)
//   ws[256..271] : csum[i]  = sum_d W[d][i]
//   ws[272..287] : h[i]     = sum_d b[d]*W[d][i]
//   ws[288]      : bb       = sum_d b[d]^2
//   ws[289]      : sb       = sum_d b[d]
// One 256-thread block; thread (i,j) owns G[i][j].
// ---------------------------------------------------------------------------
__global__ void __launch_bounds__(256)
winding_stats_kernel(const float* __restrict__ proj_w,
                     const float* __restrict__ proj_b,
                     float* __restrict__ ws) {
  const int tid = threadIdx.x;
  const int i = tid >> 4, j = tid & 15;
  float g = 0.f, cs = 0.f, hh = 0.f;
  for (int d = 0; d < EMB_D; ++d) {
    const float wi = proj_w[d * NFEAT + i];
    const float wj = proj_w[d * NFEAT + j];
    g = fmaf(wi, wj, g);
    if (j == 0) {
      cs += wi;
      hh = fmaf(proj_b[d], wi, hh);
    }
  }
  ws[tid] = g;
  if (j == 0) {
    ws[256 + i] = cs;
    ws[272 + i] = hh;
  }
  if (tid == 0) {
    float bb = 0.f, sb = 0.f;
    for (int d = 0; d < EMB_D; ++d) {
      const float b = proj_b[d];
      bb = fmaf(b, b, bb);
      sb += b;
    }
    ws[288] = bb;
    ws[289] = sb;
  }
}

// ---------------------------------------------------------------------------
// Main kernel: one block = 16 tokens x 1024 dims. 8 waves; wave w covers
// columns [w*128, w*128+128) as eight 16x16 WMMA tiles (K=16 via 4 chained
// v_wmma_f32_16x16x4_f32). LayerNorm applied per-lane using analytic stats.
// ---------------------------------------------------------------------------
__global__ void __launch_bounds__(256)
winding_embed_kernel(const int*   __restrict__ x,
                     const float* __restrict__ windings,
                     const float* __restrict__ proj_w,
                     const float* __restrict__ proj_b,
                     const float* __restrict__ gamma,
                     const float* __restrict__ beta,
                     const float* __restrict__ stats,
                     float*       __restrict__ out) {
  __shared__ float feats[16][17];   // padded: avoid LDS bank conflicts
  __shared__ float mu_s[16];
  __shared__ float rs_s[16];

  const int tile = blockIdx.x;      // 16-token tile index
  const int tid  = threadIdx.x;
  const int lane = tid & 31;
  const int wave = tid >> 5;        // 0..7

  // --- wave 0 (threads 0..15): build 16x16 feature tile + per-token stats ---
  if (tid < 16) {
    const int tok = tile * 16 + tid;
    const float t = (float)x[tok] * (1.0f / (float)VOCAB_SZ);
    const float ang0 = 6.283185307179586f * t;
    float f[NFEAT];
#pragma unroll
    for (int wd = 0; wd < 8; ++wd) {
      float s, c;
      __sincosf(ang0 * windings[wd], &s, &c);
      f[2 * wd]     = c;
      f[2 * wd + 1] = s;
      feats[tid][2 * wd]     = c;
      feats[tid][2 * wd + 1] = s;
    }
    // mu = (csum.f + sb)/D ;  E2 = (f'Gf + 2 h.f + bb)/D ; var = E2 - mu^2
    float quad = 0.f, dotc = 0.f, doth = 0.f;
#pragma unroll 4
    for (int i = 0; i < NFEAT; ++i) {
      const float fi = f[i];
      dotc = fmaf(stats[256 + i], fi, dotc);
      doth = fmaf(stats[272 + i], fi, doth);
      float acc = 0.f;
#pragma unroll
      for (int j = 0; j < NFEAT; ++j)
        acc = fmaf(stats[i * 16 + j], f[j], acc);
      quad = fmaf(fi, acc, quad);
    }
    const float bb = stats[288], sb = stats[289];
    const float mu = (dotc + sb) * (1.0f / EMB_D);
    const float e2 = (quad + 2.0f * doth + bb) * (1.0f / EMB_D);
    mu_s[tid] = mu;
    rs_s[tid] = rsqrtf((e2 - mu * mu) + LN_EPS);
  }
  __syncthreads();

  // --- A fragments: 16x4 f32 tile per k-chunk.
  // Lane l: M = l%16; VGPR0/1 hold K = koff, koff+1 where koff = (l<16 ? 0:2).
  const int mrow = lane & 15;
  const int koff = (lane >> 4) << 1;
  v2f afrag[4];
#pragma unroll
  for (int kc = 0; kc < 4; ++kc) {
    afrag[kc].x = feats[mrow][kc * 4 + koff];
    afrag[kc].y = feats[mrow][kc * 4 + koff + 1];
  }

  // C/D layout: this lane owns column (lane%16), rows mbase..mbase+7.
  const int mbase = (lane >> 4) * 8;
  float mu_r[8], rs_r[8];
#pragma unroll
  for (int r = 0; r < 8; ++r) {
    mu_r[r] = mu_s[mbase + r];
    rs_r[r] = rs_s[mbase + r];
  }

  const int row0 = tile * 16;
#pragma unroll 1
  for (int nt = 0; nt < 8; ++nt) {
    const int ncol = (wave * 8 + nt) * 16 + (lane & 15);

    // B fragment per k-chunk: B[k][n] = proj_w[ncol*16 + k]; lane owns column
    // ncol, K-pair (koff, koff+1). 8-byte aligned -> global_load_b64.
    v8f c = {};
#pragma unroll
    for (int kc = 0; kc < 4; ++kc) {
      const v2f b = *(const v2f*)(proj_w + ncol * NFEAT + kc * 4 + koff);
      c = __builtin_amdgcn_wmma_f32_16x16x4_f32(
              /*neg_a=*/false, afrag[kc], /*neg_b=*/false, b,
              /*c_mod=*/(short)0, c, /*reuse_a=*/false, /*reuse_b=*/false);
    }

    const float bias = proj_b[ncol];
    const float g    = gamma[ncol];
    const float be   = beta[ncol];
    float* op = out + (size_t)row0 * EMB_D + ncol;
#pragma unroll
    for (int r = 0; r < 8; ++r) {
      const float e = c[r] + bias;
      op[(size_t)(mbase + r) * EMB_D] = (e - mu_r[r]) * rs_r[r] * g + be;
    }
  }
}

// ---------------------------------------------------------------------------
extern "C" void kernel_launch(void* const* d_in, const int* in_sizes, int n_in,
                              void* d_out, int out_size, void* d_ws, size_t ws_size,
                              hipStream_t stream) {
  const int*   x        = (const int*)  d_in[0];  // (B,S) token ids
  const float* windings = (const float*)d_in[1];  // (8,)
  const float* proj_w   = (const float*)d_in[2];  // (1024,16) row-major
  const float* proj_b   = (const float*)d_in[3];  // (1024,)
  const float* gamma    = (const float*)d_in[4];  // (1024,)
  const float* beta     = (const float*)d_in[5];  // (1024,)
  float*       out      = (float*)d_out;          // (B,S,1024)
  float*       ws       = (float*)d_ws;           // >= 290 floats

  const int n_tok  = in_sizes[0];
  const int ntiles = n_tok / 16;

  winding_stats_kernel<<<1, 256, 0, stream>>>(proj_w, proj_b, ws);
  winding_embed_kernel<<<ntiles, 256, 0, stream>>>(x, windings, proj_w, proj_b,
                                                   gamma, beta, ws, out);
}